// PoolGraphSAGE_76063870812656
// MI455X (gfx1250) — compile-verified
//
#include <hip/hip_runtime.h>

typedef __attribute__((ext_vector_type(2))) float v2f;
typedef __attribute__((ext_vector_type(4))) float v4f;
typedef __attribute__((ext_vector_type(8))) float v8f;

#define D 128   // D_IN == D_HID == D_OUT == 128

// ---------------------------------------------------------------------------
// WMMA helper: D = A(16x4,f32) * B(4x16,f32) + C(16x16,f32), wave32.
// ---------------------------------------------------------------------------
__device__ __forceinline__ v8f wmma4(v2f a, v2f b, v8f c) {
    return __builtin_amdgcn_wmma_f32_16x16x4_f32(
        /*neg_a=*/false, a, /*neg_b=*/false, b,
        /*c_mod=*/(short)0, c, /*reuse_a=*/false, /*reuse_b=*/false);
}

__device__ __forceinline__ v4f vmax4(v4f a, v4f b) {
    v4f r;
    r.x = fmaxf(a.x, b.x); r.y = fmaxf(a.y, b.y);
    r.z = fmaxf(a.z, b.z); r.w = fmaxf(a.w, b.w);
    return r;
}

// ---------------------------------------------------------------------------
// out[M,128] = ReLU(A1[M,128]@W1[128,128] (+ A2[M,128]@W2[128,128]) + bias)
// Each wave owns ONE 16-wide output column (tn = wave&7) and marches over
// M-tiles.  The B panel for that column (32 k-steps x v2f) is preloaded into
// registers ONCE and reused for every M-tile -> steady-state loop is exactly
// one global_load_b64 (A frag) per v_wmma.
// A-frag (32-bit A 16x4): lane L holds row M=L&15, ks=2*(L>>4):
//   a.x=A[M][k+ks], a.y=A[M][k+ks+1].
// B mirrors the K split: b.x=W[k+ks][n], b.y=W[k+ks+1][n], n = tn*16+(L&15).
// C/D: VGPR r, lanes0-15 -> (M=r,N), lanes16-31 -> (M=r+8,N).
// ---------------------------------------------------------------------------
template <bool DUAL>
__global__ __launch_bounds__(256)
void gemm_relu_kernel(const float* __restrict__ A1, const float* __restrict__ W1,
                      const float* __restrict__ A2, const float* __restrict__ W2,
                      const float* __restrict__ bias, float* __restrict__ out,
                      int nrows) {
    const int lane = threadIdx.x & 31;
    const int l15  = lane & 15;
    const int hh   = lane >> 4;          // 0: K={0,1}, 1: K={2,3}
    const int ks   = 2 * hh;
    const int wid  = blockIdx.x * (blockDim.x >> 5) + (threadIdx.x >> 5);
    const int nw   = gridDim.x * (blockDim.x >> 5);   // multiple of 8
    const int tiles_m = (nrows + 15) >> 4;

    const int tn   = wid & 7;            // persistent column tile
    const int ncol = tn * 16 + l15;

    // ---- Preload B panel(s) for this column into registers (fully unrolled,
    //      constant indices -> stays in VGPRs, never scratch) ----
    v2f b1[32];
    #pragma unroll
    for (int t = 0; t < 32; ++t) {
        const float* p = W1 + (size_t)(4 * t + ks) * D + ncol;
        v2f b; b.x = p[0]; b.y = p[D];
        b1[t] = b;
    }
    v2f b2[32];
    if (DUAL) {
        #pragma unroll
        for (int t = 0; t < 32; ++t) {
            const float* p = W2 + (size_t)(4 * t + ks) * D + ncol;
            v2f b; b.x = p[0]; b.y = p[D];
            b2[t] = b;
        }
    }
    const float bv = bias[ncol];

    for (int tm = wid >> 3; tm < tiles_m; tm += (nw >> 3)) {
        const int m0 = tm * 16;
        int mrow = m0 + l15;
        if (mrow >= nrows) mrow = nrows - 1;   // clamp: keep EXEC all-ones

        v8f c = {};
        const float* a1p = A1 + (size_t)mrow * D + ks;
        #pragma unroll
        for (int t = 0; t < 32; ++t) {
            v2f a = *(const v2f*)(a1p + 4 * t);
            c = wmma4(a, b1[t], c);
        }
        if (DUAL) {
            const float* a2p = A2 + (size_t)mrow * D + ks;
            #pragma unroll
            for (int t = 0; t < 32; ++t) {
                v2f a = *(const v2f*)(a2p + 4 * t);
                c = wmma4(a, b2[t], c);
            }
        }

        #pragma unroll
        for (int r = 0; r < 8; ++r) {
            const int row = m0 + hh * 8 + r;
            if (row < nrows) {
                float v = c[r] + bv;
                out[(size_t)row * D + ncol] = v > 0.0f ? v : 0.0f;
            }
        }
    }
}

// ---------------------------------------------------------------------------
// CSR build: histogram of in-degrees, exclusive scan, index fill.
// ---------------------------------------------------------------------------
__global__ void hist_kernel(const int* __restrict__ dst, int* __restrict__ counts, int E) {
    int i = blockIdx.x * blockDim.x + threadIdx.x;
    if (i < E) atomicAdd(&counts[dst[i]], 1);
}

__global__ __launch_bounds__(1024)
void scan_kernel(const int* __restrict__ counts, int* __restrict__ row_start, int n) {
    __shared__ int sh[1024];
    __shared__ int carry;
    const int tid = threadIdx.x;
    if (tid == 0) { carry = 0; row_start[0] = 0; }
    __syncthreads();
    for (int base = 0; base < n; base += 1024) {
        const int i = base + tid;
        sh[tid] = (i < n) ? counts[i] : 0;
        __syncthreads();
        #pragma unroll
        for (int off = 1; off < 1024; off <<= 1) {
            int t = (tid >= off) ? sh[tid - off] : 0;
            __syncthreads();
            sh[tid] += t;
            __syncthreads();
        }
        const int inc   = sh[tid];
        const int cbase = carry;
        if (i < n) row_start[i + 1] = cbase + inc;
        __syncthreads();
        if (tid == 1023) carry = cbase + inc;
        __syncthreads();
    }
}

__global__ void fill_kernel(const int* __restrict__ src, const int* __restrict__ dst,
                            const int* __restrict__ row_start, int* __restrict__ cursor,
                            int* __restrict__ csr_src, int E) {
    int i = blockIdx.x * blockDim.x + threadIdx.x;
    if (i < E) {
        const int d   = dst[i];
        const int pos = row_start[d] + atomicAdd(&cursor[d], 1);
        csr_src[pos] = src[i];
    }
}

// ---------------------------------------------------------------------------
// Per-node max over gathered neighbor features (atomic-free segment_max).
// One wave per node; lane holds 4 floats -> b128 loads/stores, 512B/row.
// 4-edge unroll keeps 4 row-gathers in flight to hide L2 latency.
// Empty neighborhoods yield 0 (matches reference's isfinite->0 fixup).
// ---------------------------------------------------------------------------
__global__ __launch_bounds__(256)
void seg_max_kernel(const float* __restrict__ hpool, const int* __restrict__ row_start,
                    const int* __restrict__ csr_src, float* __restrict__ agg, int nnodes) {
    const int lane  = threadIdx.x & 31;
    const int wib   = threadIdx.x >> 5;
    const int nwave = (blockDim.x >> 5) * gridDim.x;
    for (int v = blockIdx.x * (blockDim.x >> 5) + wib; v < nnodes; v += nwave) {
        const int beg = row_start[v];
        const int end = row_start[v + 1];
        v4f acc = {0.0f, 0.0f, 0.0f, 0.0f};
        int j = beg;
        for (; j + 4 <= end; j += 4) {
            const int s0 = csr_src[j + 0];
            const int s1 = csr_src[j + 1];
            const int s2 = csr_src[j + 2];
            const int s3 = csr_src[j + 3];
            v4f m0 = ((const v4f*)(hpool + (size_t)s0 * D))[lane];
            v4f m1 = ((const v4f*)(hpool + (size_t)s1 * D))[lane];
            v4f m2 = ((const v4f*)(hpool + (size_t)s2 * D))[lane];
            v4f m3 = ((const v4f*)(hpool + (size_t)s3 * D))[lane];
            acc = vmax4(acc, vmax4(vmax4(m0, m1), vmax4(m2, m3)));
        }
        for (; j < end; ++j) {
            const int s = csr_src[j];
            v4f m = ((const v4f*)(hpool + (size_t)s * D))[lane];
            acc = vmax4(acc, m);
        }
        ((v4f*)(agg + (size_t)v * D))[lane] = acc;
    }
}

// ---------------------------------------------------------------------------
extern "C" void kernel_launch(void* const* d_in, const int* in_sizes, int n_in,
                              void* d_out, int out_size, void* d_ws, size_t ws_size,
                              hipStream_t stream) {
    const float* x   = (const float*)d_in[0];
    const int*  eidx = (const int*)  d_in[1];
    const float* Wp1 = (const float*)d_in[2];
    const float* bp1 = (const float*)d_in[3];
    const float* Wu1 = (const float*)d_in[4];
    const float* bu1 = (const float*)d_in[5];
    const float* Wp2 = (const float*)d_in[6];
    const float* bp2 = (const float*)d_in[7];
    const float* Wu2 = (const float*)d_in[8];
    const float* bu2 = (const float*)d_in[9];

    const int N = in_sizes[0] / D;
    const int E = in_sizes[1] / 2;
    const int* src = eidx;
    const int* dst = eidx + E;

    // Workspace layout (all 4-byte types): hpool | agg | h1 | counts | row_start | csr
    float* hpool  = (float*)d_ws;
    float* agg    = hpool + (size_t)N * D;
    float* h1     = agg   + (size_t)N * D;
    int*   counts = (int*)(h1 + (size_t)N * D);
    int*   rowst  = counts + N;
    int*   csr    = rowst + N + 1;

    const int eb = (E + 255) / 256;
    const int tiles_m = (N + 15) / 16;
    int gemm_blocks = tiles_m < 1024 ? tiles_m : 1024;   // 8 waves/block, 1 col each
    if (gemm_blocks < 1) gemm_blocks = 1;
    const int agg_blocks = (N + 7) / 8;

    // --- Build CSR (dst -> list of src) ---
    hipMemsetAsync(counts, 0, (size_t)N * sizeof(int), stream);
    hist_kernel<<<eb, 256, 0, stream>>>(dst, counts, E);
    scan_kernel<<<1, 1024, 0, stream>>>(counts, rowst, N);
    hipMemsetAsync(counts, 0, (size_t)N * sizeof(int), stream);   // reuse as cursor
    fill_kernel<<<eb, 256, 0, stream>>>(src, dst, rowst, counts, csr, E);

    // --- Layer 1 ---
    gemm_relu_kernel<false><<<gemm_blocks, 256, 0, stream>>>(x, Wp1, nullptr, nullptr,
                                                             bp1, hpool, N);
    seg_max_kernel<<<agg_blocks, 256, 0, stream>>>(hpool, rowst, csr, agg, N);
    gemm_relu_kernel<true><<<gemm_blocks, 256, 0, stream>>>(x, Wu1, agg, Wu1 + D * D,
                                                            bu1, h1, N);

    // --- Layer 2 ---
    gemm_relu_kernel<false><<<gemm_blocks, 256, 0, stream>>>(h1, Wp2, nullptr, nullptr,
                                                             bp2, hpool, N);
    seg_max_kernel<<<agg_blocks, 256, 0, stream>>>(hpool, rowst, csr, agg, N);
    gemm_relu_kernel<true><<<gemm_blocks, 256, 0, stream>>>(h1, Wu2, agg, Wu2 + D * D,
                                                            bu2, (float*)d_out, N);
}